// FuzzyPooling_59871844106664
// MI455X (gfx1250) — compile-verified
//
#include <hip/hip_runtime.h>

// Fuzzy pooling, 3x3 stride-3, on [16,64,192,192] f32 -> [16,64,64,64] f32.
// TRANS/VALU-bound (45 v_exp_f32 per patch); memory path is single-pass
// streaming so all global traffic uses non-temporal hints (gfx1250 TH=NT).

#define EPS_F 1e-4f
#define SIGMA_TOL_F 1e-3f
// -0.5 * log2(e): folds exp(-0.5 z^2) into a single hardware exp2.
#define NEG_HALF_LOG2E -0.72134752044448170368f

#if defined(__has_builtin)
#  if __has_builtin(__builtin_amdgcn_exp2f)
#    define FAST_EXP2(x) __builtin_amdgcn_exp2f(x)
#  else
#    define FAST_EXP2(x) exp2f(x)
#  endif
#  if __has_builtin(__builtin_amdgcn_rcpf)
#    define FAST_RCP(x) __builtin_amdgcn_rcpf(x)
#  else
#    define FAST_RCP(x) (1.0f / (x))
#  endif
#else
#  define FAST_EXP2(x) exp2f(x)
#  define FAST_RCP(x) (1.0f / (x))
#endif

__global__ __launch_bounds__(256) void fuzzy_pool_kernel(
    const float* __restrict__ x, float* __restrict__ y, int total)
{
    int t = blockIdx.x * 256 + threadIdx.x;
    if (t >= total) return;

    // Fixed geometry: B*C = 1024, O = 64, H = W = 192.
    constexpr int W  = 192;
    constexpr int HW = 192 * 192;

    const int ox = t & 63;
    const int oy = (t >> 6) & 63;
    const int bc = t >> 12;

    const float* base = x + (size_t)bc * HW + (size_t)(oy * 3) * W + (size_t)(ox * 3);

    // Load the 3x3 patch (row-major n = r*3+c). Each row is 12 contiguous
    // bytes; a wave32 reads 384 contiguous bytes per row -> fully coalesced.
    float p[9];
#pragma unroll
    for (int r = 0; r < 3; ++r) {
        const float* rp = base + r * W;
        p[r * 3 + 0] = __builtin_nontemporal_load(rp + 0);
        p[r * 3 + 1] = __builtin_nontemporal_load(rp + 1);
        p[r * 3 + 2] = __builtin_nontemporal_load(rp + 2);
    }

    // Nested partial sums: 8 adds total.
    const float s3 = p[3] + p[4] + p[5];
    const float s5 = s3 + p[2] + p[6];
    const float s7 = s5 + p[1] + p[7];
    const float s9 = s7 + p[0] + p[8];

    // kmm = [mean p[1:8], mean p[2:7], mean p[3:6], p[4], mean p[0:9]]
    float kmm[5];
    kmm[0] = s7 * (1.0f / 7.0f);
    kmm[1] = s5 * 0.2f;
    kmm[2] = s3 * (1.0f / 3.0f);
    kmm[3] = p[4];
    kmm[4] = s9 * (1.0f / 9.0f);

    const float v_avg = (kmm[0] + kmm[1] + kmm[2] + kmm[3] + kmm[4]) * 0.2f;

    // sigma0 = mean |p - v_avg|
    float so = 0.0f;
#pragma unroll
    for (int n = 0; n < 9; ++n) so += fabsf(p[n] - v_avg);
    const float sigma0 = so * (1.0f / 9.0f);

    // sigma[j] = kmm[j] + EPS for j<4 (same windows), sigma[4] = sigma0 + EPS.
    float sig[5];
    sig[0] = kmm[0] + EPS_F;
    sig[1] = kmm[1] + EPS_F;
    sig[2] = kmm[2] + EPS_F;
    sig[3] = kmm[3] + EPS_F;
    sig[4] = sigma0 + EPS_F;

    // Fold per-j constants: pi = exp2(d*d * c2[j]), c2[j] = -0.5*log2e / sig^2.
    float c2[5];
#pragma unroll
    for (int j = 0; j < 5; ++j) {
        const float r = FAST_RCP(sig[j]);
        c2[j] = NEG_HALF_LOG2E * r * r;
    }

    // 9x5 membership grid: thresh = min_n max_j pi, avg_pi[n] = mean_j pi.
    float thresh = 3.402823466e38f;
    float num = 0.0f, den = 0.0f, avg4 = 0.0f;
#pragma unroll
    for (int n = 0; n < 9; ++n) {
        float mx = 0.0f;  // pi >= 0 always
        float sm = 0.0f;
#pragma unroll
        for (int j = 0; j < 5; ++j) {
            const float d = p[n] - kmm[j];
            const float e = FAST_EXP2(d * d * c2[j]);
            mx = fmaxf(mx, e);
            sm += e;
        }
        thresh = fminf(thresh, mx);
        const float a = 0.2f * sm;
        if (n == 4) avg4 = a;
        num = fmaf(a, p[n], num);
        den += a;
    }

    const float mean9    = kmm[4];
    const float denoised = num / den;  // once per thread: keep IEEE division

    const bool primary   = (avg4 >= thresh);
    const bool secondary = (sig[4] < SIGMA_TOL_F);

    const float result = primary ? mean9 : (secondary ? v_avg : denoised);
    __builtin_nontemporal_store(result, y + t);
}

extern "C" void kernel_launch(void* const* d_in, const int* in_sizes, int n_in,
                              void* d_out, int out_size, void* d_ws, size_t ws_size,
                              hipStream_t stream) {
    (void)in_sizes; (void)n_in; (void)d_ws; (void)ws_size;
    const float* x = (const float*)d_in[0];
    float* y = (float*)d_out;

    const int total = out_size;  // 16*64*64*64 = 4,194,304
    const int block = 256;       // 8 wave32s per block
    const int grid  = (total + block - 1) / block;
    fuzzy_pool_kernel<<<grid, block, 0, stream>>>(x, y, total);
}